// ChamferLoss_47682726920370
// MI455X (gfx1250) — compile-verified
//
#include <hip/hip_runtime.h>

// ---------------------------------------------------------------------------
// Chamfer distance on MI455X (gfx1250).
//   d2(a_i, b_j) = |a_i|^2 + |b_j|^2 - 2 a_i.b_j
// The a.b term for a 16x16 tile of point pairs is one V_WMMA_F32_16X16X4_F32
// (K padded 3 -> 4 with zeros).  One wave32 owns 16 rows of A and sweeps all
// columns of B, keeping 8 per-lane running-min VGPRs (the 16x16 f32 C layout).
// Working set (1.5 MB) is resident in the 192 MB L2, so no prefetch needed;
// the loop body is 1 b96 load + 1 WMMA + ~15 packed VALU ops per tile.
// ---------------------------------------------------------------------------

typedef __attribute__((ext_vector_type(2))) float v2f;
typedef __attribute__((ext_vector_type(8))) float v8f;

#define WAVES_PER_BLOCK 8
#define TILE 16

__global__ __launch_bounds__(256) void chamfer_side_kernel(
    const float* __restrict__ A,    // [batch, N, 3] query points (row side)
    const float* __restrict__ Bp,   // [batch, M, 3] target points (col side)
    float* __restrict__ partials,   // [B * N/16] per-tile sums of row-mins
    int N, int M)
{
  const int lane = threadIdx.x & 31;
  const int wave = threadIdx.x >> 5;
  const int l15  = lane & 15;       // column / row-in-tile index
  const int half = lane >> 4;       // 0: K=0..1 lanes, 1: K=2..3 lanes

  const int tilesPerBatch = N / TILE;
  const int tileGlobal = blockIdx.x * WAVES_PER_BLOCK + wave;
  const int batch   = tileGlobal / tilesPerBatch;
  const int tileRow = tileGlobal - batch * tilesPerBatch;

  const float* ap = A  + ((size_t)batch * N + (size_t)tileRow * TILE) * 3;
  const float* bp = Bp +  (size_t)batch * M * 3;

  // ---- A operand (16x4 f32 -> 2 VGPRs). Lane l15 carries row l15.
  //      lanes 0-15: VGPR0=K0(x), VGPR1=K1(y); lanes 16-31: VGPR0=K2(z), VGPR1=K3(0)
  const float ax = ap[3 * l15 + 0];
  const float ay = ap[3 * l15 + 1];
  const float az = ap[3 * l15 + 2];
  v2f Aop;
  Aop.x = half ? az : ax;
  Aop.y = half ? 0.0f : ay;

  // ---- |a|^2 per C-layout row: VGPR r holds row (r + 8*half)
  float a2r[8];
#pragma unroll
  for (int r = 0; r < 8; ++r) {
    const int m = r + 8 * half;
    const float x = ap[3 * m + 0];
    const float y = ap[3 * m + 1];
    const float z = ap[3 * m + 2];
    a2r[r] = x * x + y * y + z * z;
  }

  float mn[8];
#pragma unroll
  for (int r = 0; r < 8; ++r) mn[r] = 3.402823466e+38f;

  // ---- sweep all column tiles of B; EXEC stays all-ones (WMMA requirement)
  for (int j = 0; j < M; j += TILE) {
    const int col = j + l15;
    const float bx = bp[3 * col + 0];
    const float by = bp[3 * col + 1];
    const float bz = bp[3 * col + 2];

    v2f Bop;                                        // B operand (4x16 f32)
    Bop.x = half ? bz : bx;
    Bop.y = half ? 0.0f : by;
    const float b2 = bx * bx + by * by + bz * bz;

    v8f c = {};
    // D = A(16x4) x B(4x16) + 0   ->  v_wmma_f32_16x16x4_f32
    c = __builtin_amdgcn_wmma_f32_16x16x4_f32(
        /*neg_a=*/false, Aop, /*neg_b=*/false, Bop,
        /*c_mod=*/(short)0, c, /*reuse_a=*/false, /*reuse_b=*/false);

#pragma unroll
    for (int r = 0; r < 8; ++r) {
      const float d = a2r[r] + b2 - 2.0f * c[r];
      mn[r] = fminf(mn[r], d);
    }
  }

  // ---- row-min across the 16 lanes of each half (wave32 xor shuffles)
#pragma unroll
  for (int mask = 1; mask <= 8; mask <<= 1) {
#pragma unroll
    for (int r = 0; r < 8; ++r)
      mn[r] = fminf(mn[r], __shfl_xor(mn[r], mask, 32));
  }

  // ---- pick one lane per row: lane r -> row r, lane 16+r -> row r+8,
  //      clamp at zero (clamp commutes with min), sum 16 row-mins of the tile
  float s = 0.0f;
#pragma unroll
  for (int r = 0; r < 8; ++r)
    s = (l15 == r) ? fmaxf(mn[r], 0.0f) : s;

#pragma unroll
  for (int mask = 16; mask >= 1; mask >>= 1)
    s += __shfl_xor(s, mask, 32);

  if (lane == 0) partials[tileGlobal] = s;
}

// Deterministic final reduction: one block tree-reduces all tile partials.
__global__ __launch_bounds__(256) void chamfer_reduce_kernel(
    const float* __restrict__ partials, float* __restrict__ out,
    int P, float scale)
{
  __shared__ float sm[256];
  float s = 0.0f;
  for (int i = threadIdx.x; i < P; i += 256) s += partials[i];
  sm[threadIdx.x] = s;
  __syncthreads();
#pragma unroll
  for (int off = 128; off > 0; off >>= 1) {
    if ((int)threadIdx.x < off) sm[threadIdx.x] += sm[threadIdx.x + off];
    __syncthreads();
  }
  if (threadIdx.x == 0) out[0] = sm[0] * scale;
}

extern "C" void kernel_launch(void* const* d_in, const int* in_sizes, int n_in,
                              void* d_out, int out_size, void* d_ws, size_t ws_size,
                              hipStream_t stream) {
  (void)in_sizes; (void)n_in; (void)out_size; (void)ws_size;
  const float* a = (const float*)d_in[0];   // predict_pc [8,8192,3] f32
  const float* b = (const float*)d_in[1];   // gt_pc      [8,8192,3] f32
  float* out = (float*)d_out;               // scalar f32
  float* ws  = (float*)d_ws;

  const int B = 8, N = 8192, M = 8192;
  const int tilesPerDir = B * (N / TILE);               // 4096 waves / direction
  const int blocks = tilesPerDir / WAVES_PER_BLOCK;     // 512 blocks of 256 thr

  // direction 1: for each a, nearest b
  chamfer_side_kernel<<<blocks, 256, 0, stream>>>(a, b, ws, N, M);
  // direction 2: for each b, nearest a
  chamfer_side_kernel<<<blocks, 256, 0, stream>>>(b, a, ws + tilesPerDir, M, N);
  // (sum1 + sum2) / (2 * B * N)
  chamfer_reduce_kernel<<<1, 256, 0, stream>>>(
      ws, out, 2 * tilesPerDir, 1.0f / (2.0f * (float)B * (float)N));
}